// JittorDeformNet_35373350650541
// MI455X (gfx1250) — compile-verified
//
#include <hip/hip_runtime.h>
#include <hip/hip_bf16.h>

typedef __attribute__((ext_vector_type(2))) float v2f;
typedef __attribute__((ext_vector_type(8))) float v8f;

#define IH 28
#define IW 28
#define HP 30
#define WP 30
#define NPIX (IH * IW)      // 784
#define NFEAT 1568          // 8 * 14 * 14

// ---------------------------------------------------------------------------
// Kernel 1: fused offset-conv + deformable bilinear sampling + stride-3 conv
//           + relu + 2x2 maxpool.  One block per image.
// ---------------------------------------------------------------------------
__global__ __launch_bounds__(256) void deform_feat_kernel(
    const float* __restrict__ x,       // (B,1,28,28)
    const float* __restrict__ w_off,   // (18,1,3,3)
    const float* __restrict__ b_off,   // (18)
    const float* __restrict__ w_conv,  // (8,1,3,3)
    float* __restrict__ feat)          // (B,1568)
{
    __shared__ float xpad[HP * WP];      // padded image
    __shared__ float convb[8 * NPIX];    // conv output (pre-pool)
    __shared__ float swoff[18 * 9];
    __shared__ float swc[8 * 9];
    __shared__ float sboff[18];

    const int b   = blockIdx.x;
    const int tid = threadIdx.x;
    const float* xb = x + b * NPIX;

    if (tid < 162) swoff[tid] = w_off[tid];
    if (tid < 72)  swc[tid]   = w_conv[tid];
    if (tid < 18)  sboff[tid] = b_off[tid];

    // build zero-padded 30x30 image in one pass
    for (int i = tid; i < HP * WP; i += 256) {
        int r = i / WP, c = i % WP;
        bool in = (r >= 1) & (r <= IH) & (c >= 1) & (c <= IW);
        xpad[i] = in ? xb[(r - 1) * IW + (c - 1)] : 0.0f;
    }
    __syncthreads();

    for (int p = tid; p < NPIX; p += 256) {
        const int h = p / IW, w = p % IW;

        // ---- offset conv (pad=1), 18 channels ----
        float off[18];
        #pragma unroll
        for (int c = 0; c < 18; ++c) off[c] = sboff[c];
        #pragma unroll
        for (int di = 0; di < 3; ++di) {
            #pragma unroll
            for (int dj = 0; dj < 3; ++dj) {
                const float xv = xpad[(h + di) * WP + (w + dj)];
                #pragma unroll
                for (int c = 0; c < 18; ++c)
                    off[c] = fmaf(swoff[c * 9 + di * 3 + dj], xv, off[c]);
            }
        }

        // ---- 9 bilinear samples + fused stride-3 conv ----
        float acc[8];
        #pragma unroll
        for (int oc = 0; oc < 8; ++oc) acc[oc] = 0.0f;

        #pragma unroll
        for (int k = 0; k < 9; ++k) {
            const float pnx = (float)(k / 3 - 1);
            const float pny = (float)(k % 3 - 1);
            const float pxr = (float)(h + 1) + pnx + off[k];      // padded coords
            const float pyr = (float)(w + 1) + pny + off[k + 9];

            const float qx = floorf(pxr);                          // floor of UNclipped p
            const float qy = floorf(pyr);
            const float qltx = fminf(fmaxf(qx,        0.0f), (float)(HP - 1));
            const float qlty = fminf(fmaxf(qy,        0.0f), (float)(WP - 1));
            const float qrbx = fminf(fmaxf(qx + 1.0f, 0.0f), (float)(HP - 1));
            const float qrby = fminf(fmaxf(qy + 1.0f, 0.0f), (float)(WP - 1));
            const float px = fminf(fmaxf(pxr, 0.0f), (float)(HP - 1));
            const float py = fminf(fmaxf(pyr, 0.0f), (float)(WP - 1));

            const float g_lt = (1.0f + (qltx - px)) * (1.0f + (qlty - py));
            const float g_rb = (1.0f - (qrbx - px)) * (1.0f - (qrby - py));
            const float g_lb = (1.0f + (qltx - px)) * (1.0f - (qrby - py));
            const float g_rt = (1.0f - (qrbx - px)) * (1.0f + (qlty - py));

            const int ltx = (int)qltx, lty = (int)qlty;
            const int rbx = (int)qrbx, rby = (int)qrby;
            const float xlt = xpad[ltx * WP + lty];
            const float xrb = xpad[rbx * WP + rby];
            const float xlb = xpad[ltx * WP + rby];
            const float xrt = xpad[rbx * WP + lty];

            const float s = g_lt * xlt + g_rb * xrb + g_lb * xlb + g_rt * xrt;

            #pragma unroll
            for (int oc = 0; oc < 8; ++oc)
                acc[oc] = fmaf(swc[oc * 9 + k], s, acc[oc]);
        }
        #pragma unroll
        for (int oc = 0; oc < 8; ++oc)
            convb[oc * NPIX + p] = fmaxf(acc[oc], 0.0f);           // relu
    }
    __syncthreads();

    // ---- 2x2 maxpool -> (8,14,14) flat features ----
    for (int i = tid; i < NFEAT; i += 256) {
        const int c  = i / 196;
        const int r  = i % 196;
        const int ph = r / 14, pw = r % 14;
        const int base = c * NPIX + (ph * 2) * IW + pw * 2;
        const float m = fmaxf(fmaxf(convb[base],      convb[base + 1]),
                              fmaxf(convb[base + IW], convb[base + IW + 1]));
        feat[(size_t)b * NFEAT + i] = m;
    }
}

// ---------------------------------------------------------------------------
// Kernel 2: FC layer  (B x 1568) @ (1568 x 10)^T + bias  via V_WMMA_F32_16X16X4_F32
// Zero-padded 16x1568 B matrix staged in LDS once per block -> unconditional
// ds_load_b64 for B fragments (no exec-mask churn in the inner loop).
// ---------------------------------------------------------------------------
__global__ __launch_bounds__(256) void fc_wmma_kernel(
    const float* __restrict__ feat,   // (B,1568)
    const float* __restrict__ w_fc,   // (10,1568)
    const float* __restrict__ b_fc,   // (10)
    float* __restrict__ out,          // (B,10)
    int Btotal)
{
    __shared__ float Bs[16 * NFEAT];  // 98 KB: rows 0..9 = w_fc, rows 10..15 = 0

    const int tid = threadIdx.x;
    for (int i = tid; i < 16 * NFEAT; i += 256) {
        const int row = i / NFEAT;
        Bs[i] = (row < 10) ? w_fc[i] : 0.0f;
    }
    __syncthreads();

    const int lane = tid & 31;
    const int wave = (blockIdx.x * blockDim.x + tid) >> 5;
    const int row0 = wave * 16;
    if (row0 >= Btotal) return;

    const int m  = lane & 15;          // A: row index / B,C: column index
    const int kh = lane >> 4;          // 0: K={0,1}, 1: K={2,3}

    const float* arow = feat + (size_t)(row0 + m) * NFEAT + 2 * kh;
    const float* brow = Bs + m * NFEAT + 2 * kh;   // 8-byte aligned

    v8f c = {};
    #pragma unroll 4
    for (int k = 0; k < NFEAT; k += 4) {
        const v2f a  = *(const v2f*)(arow + k);    // global_load_b64
        const v2f bb = *(const v2f*)(brow + k);    // ds_load_b64 (zeros for n>=10)
        c = __builtin_amdgcn_wmma_f32_16x16x4_f32(
                /*neg_a=*/false, a, /*neg_b=*/false, bb,
                /*c_mod=*/(short)0, c, /*reuse_a=*/false, /*reuse_b=*/false);
    }

    // C/D layout: VGPR r -> lanes 0-15: M=r, N=lane; lanes 16-31: M=r+8, N=lane-16
    const int n = m;
    if (n < 10) {
        const float bias = b_fc[n];
        const int mbase = row0 + (kh ? 8 : 0);
        #pragma unroll
        for (int r = 0; r < 8; ++r)
            out[(size_t)(mbase + r) * 10 + n] = c[r] + bias;
    }
}

// ---------------------------------------------------------------------------
extern "C" void kernel_launch(void* const* d_in, const int* in_sizes, int n_in,
                              void* d_out, int out_size, void* d_ws, size_t ws_size,
                              hipStream_t stream) {
    const float* x      = (const float*)d_in[0];
    const float* w_off  = (const float*)d_in[1];
    const float* b_off  = (const float*)d_in[2];
    const float* w_conv = (const float*)d_in[3];
    const float* w_fc   = (const float*)d_in[4];
    const float* b_fc   = (const float*)d_in[5];
    float* out  = (float*)d_out;
    float* feat = (float*)d_ws;                 // needs B*1568*4 = ~25.7 MB

    const int B = in_sizes[0] / NPIX;           // 4096

    deform_feat_kernel<<<B, 256, 0, stream>>>(x, w_off, b_off, w_conv, feat);

    const int waves  = (B + 15) / 16;           // 256
    const int blocks = (waves + 7) / 8;         // 32 blocks x 8 waves
    fc_wmma_kernel<<<blocks, 256, 0, stream>>>(feat, w_fc, b_fc, out, B);
}